// _CustomLinear_44152263803135
// MI455X (gfx1250) — compile-verified
//
#include <hip/hip_runtime.h>

// ---------------------------------------------------------------------------
// y[t,o] = sum_k x[t,k] * dequant(w)[o,k]
//   dequant per (o, group of 8 k): scale = max(maxabs/7.5, eps),
//   wq = clip(rint(w/scale), -8, 7) * scale      (all f32, like reference)
//
// MI455X: HBM-bound (weight = 180MB @ 23.3TB/s ~= 7.7us floor). Strategy:
// stream weights once (non-temporal), fuse dequant in-lane (8 consecutive K
// per lane == one quant group -> no cross-lane ops), contract with
// V_WMMA_F32_16X16X4_F32 (f32-exact matrix op). K order fed to WMMA is a
// custom permutation applied identically to A and B (dot products are
// K-permutation invariant). 16 waves/block + unroll 2 keeps ~4KB of loads in
// flight per wave to cover HBM latency at full bandwidth.
// ---------------------------------------------------------------------------

typedef __attribute__((ext_vector_type(2))) float v2f;
typedef __attribute__((ext_vector_type(4))) float v4f;
typedef __attribute__((ext_vector_type(8))) float v8f;

constexpr int O_TOTAL = 11008;
constexpr int K_TOTAL = 4096;
constexpr int WAVES   = 16;                // 512 threads/block
constexpr int KCHUNK  = K_TOTAL / WAVES;   // 256 -> 16 iterations/wave
constexpr float QEPS  = 1.1920929e-07f;

__device__ __forceinline__ float dq1(float v, float rscale, float scale) {
    float r = __builtin_rintf(v * rscale);        // v_rndne: round-half-even
    r = fminf(fmaxf(r, -8.0f), 7.0f);             // -> single v_med3_num_f32
    return r * scale;
}

__global__ __launch_bounds__(512)
void w4g8_gemm_wmma(const float* __restrict__ x,
                    const float* __restrict__ w,
                    float* __restrict__ y)
{
    const int tid  = threadIdx.x;
    const int wave = tid >> 5;
    const int lane = tid & 31;
    const int n    = lane & 15;        // A row (token t) and B col (output o)
    const int h    = lane >> 4;        // half-wave: owns logical K slots {2h, 2h+1}
    const int o0   = blockIdx.x * 16;

    const float* xrow = x + (size_t)n * K_TOTAL;            // t = n
    const float* wrow = w + (size_t)(o0 + n) * K_TOTAL;     // o = o0 + n

    v8f acc = {};

    const int kbeg = wave * KCHUNK;
    const int kend = kbeg + KCHUNK;

    #pragma unroll 2
    for (int k0 = kbeg; k0 < kend; k0 += 16) {
        const int kb = k0 + h * 8;     // this lane's 8 consecutive K = 1 group

        // x tile: same per-lane K mapping as B so the contraction lines up
        v4f xa = *(const v4f*)(xrow + kb);
        v4f xb = *(const v4f*)(xrow + kb + 4);
        // weight tile: streamed once -> non-temporal (keep x resident in L2)
        v4f wa = __builtin_nontemporal_load((const v4f*)(wrow + kb));
        v4f wb = __builtin_nontemporal_load((const v4f*)(wrow + kb + 4));

        // ---- in-lane group quant/dequant (group of 8 == this lane's span) -
        float ma = fmaxf(fmaxf(fabsf(wa.x), fabsf(wa.y)),
                         fmaxf(fabsf(wa.z), fabsf(wa.w)));
        float mb = fmaxf(fmaxf(fabsf(wb.x), fabsf(wb.y)),
                         fmaxf(fabsf(wb.z), fabsf(wb.w)));
        float maxabs = fmaxf(ma, mb);
        float scale  = fmaxf(maxabs / 7.5f, QEPS);   // IEEE div: value reaches y
        float rscale = __builtin_amdgcn_rcpf(scale); // 1-ulp rcp: only feeds rint

        float q0 = dq1(wa.x, rscale, scale);
        float q1 = dq1(wa.y, rscale, scale);
        float q2 = dq1(wa.z, rscale, scale);
        float q3 = dq1(wa.w, rscale, scale);
        float q4 = dq1(wb.x, rscale, scale);
        float q5 = dq1(wb.y, rscale, scale);
        float q6 = dq1(wb.z, rscale, scale);
        float q7 = dq1(wb.w, rscale, scale);

        // ---- 4x V_WMMA_F32_16X16X4_F32, physical K pairs (2i, 2i+1) -------
        v2f a, b;
        a.x = xa.x; a.y = xa.y; b.x = q0; b.y = q1;
        acc = __builtin_amdgcn_wmma_f32_16x16x4_f32(false, a, false, b,
                                                    (short)0, acc, false, false);
        a.x = xa.z; a.y = xa.w; b.x = q2; b.y = q3;
        acc = __builtin_amdgcn_wmma_f32_16x16x4_f32(false, a, false, b,
                                                    (short)0, acc, false, false);
        a.x = xb.x; a.y = xb.y; b.x = q4; b.y = q5;
        acc = __builtin_amdgcn_wmma_f32_16x16x4_f32(false, a, false, b,
                                                    (short)0, acc, false, false);
        a.x = xb.z; a.y = xb.w; b.x = q6; b.y = q7;
        acc = __builtin_amdgcn_wmma_f32_16x16x4_f32(false, a, false, b,
                                                    (short)0, acc, false, false);
    }

    // ---- cross-wave reduction of the 16x16 partial tiles in LDS -----------
    __shared__ float red[WAVES * 256];
    #pragma unroll
    for (int v = 0; v < 8; ++v)
        red[wave * 256 + v * 32 + lane] = acc[v];
    __syncthreads();

    if (tid < 256) {
        float s = 0.0f;
        #pragma unroll
        for (int wv = 0; wv < WAVES; ++wv)
            s += red[wv * 256 + tid];

        // C layout: VGPR v, lanes 0-15 -> (M=v, N=lane); lanes 16-31 -> M=v+8
        const int v  = tid >> 5;
        const int l  = tid & 31;
        const int m  = v + 8 * (l >> 4);   // token t
        const int nn = l & 15;             // output col within tile
        y[(size_t)m * O_TOTAL + o0 + nn] = s;
    }
}

extern "C" void kernel_launch(void* const* d_in, const int* in_sizes, int n_in,
                              void* d_out, int out_size, void* d_ws, size_t ws_size,
                              hipStream_t stream) {
    const float* x = (const float*)d_in[0];   // [1,16,4096]
    const float* w = (const float*)d_in[1];   // [11008,4096]
    float* y = (float*)d_out;                 // [1,16,11008]
    (void)in_sizes; (void)n_in; (void)out_size; (void)d_ws; (void)ws_size;

    dim3 grid(O_TOTAL / 16);                  // 688 tiles of 16 outputs
    dim3 block(512);                          // 16 waves (wave32)
    w4g8_gemm_wmma<<<grid, block, 0, stream>>>(x, w, y);
}